// GTLModuleV1_55997783605457
// MI455X (gfx1250) — compile-verified
//
#include <hip/hip_runtime.h>
#include <stdint.h>

#define GLOBAL_AS __attribute__((address_space(1)))
#define LDS_AS    __attribute__((address_space(3)))

#ifndef __has_builtin
#define __has_builtin(x) 0
#endif

typedef int v4i_ __attribute__((vector_size(16)));        // async builtin param type
typedef unsigned int u32x4 __attribute__((vector_size(16)));  // TDM D# group 0
typedef int i32x8 __attribute__((vector_size(32)));           // TDM D# group 1 / ext
typedef int i32x4 __attribute__((vector_size(16)));           // TDM D# groups 2/3

static constexpr int Bb = 8, Cc = 144, Nn = 4096, Kk = 16, Gg = 9, Dd = 16;
static constexpr int BG   = Bb * Gg;        // 72
static constexpr int FEAT = Bb * Cc * Nn;   // 4718592
static constexpr int CENT = BG * Nn;        // 294912

// ---------------------------------------------------------------- zero pass
__global__ __launch_bounds__(256) void zero_cent(float* __restrict__ cent) {
  int i = blockIdx.x * 256 + threadIdx.x;
  if (i < CENT) cent[i] = 0.f;
}

// ------------------------------------------- transpose (bg,d,N) -> (bg,N,d)
// Tile fetch (16 rows x 256 floats, row stride N) is a single TDM 2D
// descriptor: DMA engine deposits the tile into LDS *with* the +1 DWORD row
// padding (pad_interval=7 -> every 256 DWORDs) needed for conflict-free
// column reads. Makes every later neighbor gather one 64B line.
__global__ __launch_bounds__(256) void transpose_dn(const float* __restrict__ q,
                                                    const float* __restrict__ v,
                                                    float* __restrict__ qt,
                                                    float* __restrict__ vt) {
  const int chunks = Nn / 256;
  const int bg = blockIdx.x / chunks;
  const int n0 = (blockIdx.x % chunks) * 256;
  const float* __restrict__ src = blockIdx.y ? v : q;
  float* __restrict__       dst = blockIdx.y ? vt : qt;

  __shared__ float tile[Dd][256 + 1];
  const int t = threadIdx.x;

#if __has_builtin(__builtin_amdgcn_tensor_load_to_lds) && \
    __has_builtin(__builtin_amdgcn_s_wait_tensorcnt)
  if (t < 32) {  // one wave issues the DMA; TENSORcnt tracks completion
    const uint64_t ga = (uint64_t)(uintptr_t)(src + (size_t)(bg * Dd) * Nn + n0);
    const uint32_t la = (uint32_t)(uintptr_t)(&tile[0][0]);
    // D# group0: count=1 | lds_addr | global_addr | type=2
    u32x4 g0 = { 1u, la, (uint32_t)ga, (uint32_t)(ga >> 32) | 0x80000000u };
    // D# group1: data_size=4B, pad_enable, pad_interval=256dw, pad_amount=1dw,
    //            tensor_dim0=4096, tensor_dim1=16, tile_dim0=256, tile_dim1=16,
    //            tensor_dim0_stride=4096
    i32x8 g1 = { 0x01D20000, 0x10000000, 0x00100000, 0x01000000, 16, 4096, 0, 0 };
    i32x4 gz = { 0, 0, 0, 0 };
    i32x8 gz8 = { 0, 0, 0, 0, 0, 0, 0, 0 };
    __builtin_amdgcn_tensor_load_to_lds(g0, g1, gz, gz, gz8, 0);
    __builtin_amdgcn_s_wait_tensorcnt(0);
  }
#else
#pragma unroll
  for (int d = 0; d < Dd; ++d)
    tile[d][t] = src[(bg * Dd + d) * Nn + n0 + t];   // coalesced along n
#endif
  __syncthreads();

  const int base = (bg * Nn + n0) * Dd;              // contiguous 16KB out tile
#pragma unroll
  for (int s = 0; s < Dd; ++s) {
    int i = s * 256 + t;                             // d = i&15, n' = i>>4
    dst[base + i] = tile[i & (Dd - 1)][i >> 4];      // coalesced stores
  }
}

// ------------------------------------------------------------- fused kernel
// block = 256 threads = 16 half-wave groups; one group per point n.
// Lane role = neighbor k: per-lane 4x b128 gathers, in-lane d-dot (16 FMA),
// 8 width-16 shuffles for softmax, 1 exp/lane, direct centrality atomic,
// k-reduction + output transpose through LDS for coalesced 64B stores.
__global__ __launch_bounds__(256) void attn_fused(const float* __restrict__ qt,
                                                  const float* __restrict__ vt,
                                                  const int* __restrict__ idx,
                                                  float* __restrict__ feat,
                                                  float* __restrict__ cent) {
  const int bg  = blockIdx.y;
  const int b   = bg / Gg;
  const int n0  = blockIdx.x * 16;
  const int t   = threadIdx.x;
  const int grp = t >> 4;   // which n within the tile
  const int ln  = t & 15;   // neighbor slot k

  __shared__ int   s_idx[256];                        // [grp][k]
  __shared__ __align__(16) float s_q[256];            // [grp][d]
  __shared__ __align__(16) float s_mat[16][16][20];   // [grp][k][d], 20 keeps rows 16B-aligned
  __shared__ float s_out[Dd][17];                     // output transpose staging

  const int idx_base = (b  * Nn + n0) * Kk;   // 256 consecutive ints
  const int q_base   = (bg * Nn + n0) * Dd;   // 256 consecutive floats

#if __has_builtin(__builtin_amdgcn_global_load_async_to_lds_b128) && \
    __has_builtin(__builtin_amdgcn_s_wait_asynccnt)
  // CDNA5 async bulk copy: 128 lanes x 16B stage both tiles (ASYNCcnt path)
  if (t < 64) {
    GLOBAL_AS v4i_* g = (GLOBAL_AS v4i_*)(uintptr_t)(idx + idx_base + t * 4);
    LDS_AS    v4i_* l = (LDS_AS    v4i_*)(uintptr_t)(&s_idx[t * 4]);
    __builtin_amdgcn_global_load_async_to_lds_b128(g, l, 0, 0);
  } else if (t < 128) {
    const int u = t - 64;
    GLOBAL_AS v4i_* g = (GLOBAL_AS v4i_*)(uintptr_t)(qt + q_base + u * 4);
    LDS_AS    v4i_* l = (LDS_AS    v4i_*)(uintptr_t)(&s_q[u * 4]);
    __builtin_amdgcn_global_load_async_to_lds_b128(g, l, 0, 0);
  }
  __builtin_amdgcn_s_wait_asynccnt(0);
#else
  s_idx[t] = idx[idx_base + t];
  s_q[t]   = qt[q_base + t];
#endif
  __syncthreads();

  // ---- per-lane neighbor gather: one 64B line each for key and value
  const int myj = s_idx[grp * 16 + ln];
  const float4* kp = (const float4*)(qt + ((size_t)bg * Nn + myj) * Dd);
  float4 k0 = kp[0], k1 = kp[1], k2 = kp[2], k3 = kp[3];
  const float4* vp = (const float4*)(vt + ((size_t)bg * Nn + myj) * Dd);
  float4 v0 = vp[0], v1 = vp[1], v2 = vp[2], v3 = vp[3];

  // q vector for this point: LDS broadcast reads (same address per group)
  const float4* qp = (const float4*)(s_q + grp * 16);
  float4 q0 = qp[0], q1 = qp[1], q2 = qp[2], q3 = qp[3];

  // ---- attention logit: in-lane 16-term dot product (no shuffles)
  float p;
  p = q0.x * k0.x;
  p = fmaf(q0.y, k0.y, p); p = fmaf(q0.z, k0.z, p); p = fmaf(q0.w, k0.w, p);
  p = fmaf(q1.x, k1.x, p); p = fmaf(q1.y, k1.y, p); p = fmaf(q1.z, k1.z, p); p = fmaf(q1.w, k1.w, p);
  p = fmaf(q2.x, k2.x, p); p = fmaf(q2.y, k2.y, p); p = fmaf(q2.z, k2.z, p); p = fmaf(q2.w, k2.w, p);
  p = fmaf(q3.x, k3.x, p); p = fmaf(q3.y, k3.y, p); p = fmaf(q3.z, k3.z, p); p = fmaf(q3.w, k3.w, p);

  // ---- softmax across the 16 neighbor lanes: max + sum butterflies
  float m = p;
  m = fmaxf(m, __shfl_xor(m, 1, 16));
  m = fmaxf(m, __shfl_xor(m, 2, 16));
  m = fmaxf(m, __shfl_xor(m, 4, 16));
  m = fmaxf(m, __shfl_xor(m, 8, 16));
  float e = __expf(p - m);                  // exactly one exp per lane
  float s = e;
  s += __shfl_xor(s, 1, 16);
  s += __shfl_xor(s, 2, 16);
  s += __shfl_xor(s, 4, 16);
  s += __shfl_xor(s, 8, 16);
  const float w = e / s;                    // normalized attention weight

  // ---- centrality scatter-add: lane k owns neighbor k
  atomicAdd(cent + bg * Nn + myj, w);

  // ---- weighted value row -> LDS tile (row k of this group's 16x16 matrix)
  float4* row = (float4*)&s_mat[grp][ln][0];
  row[0] = make_float4(w * v0.x, w * v0.y, w * v0.z, w * v0.w);
  row[1] = make_float4(w * v1.x, w * v1.y, w * v1.z, w * v1.w);
  row[2] = make_float4(w * v2.x, w * v2.y, w * v2.z, w * v2.w);
  row[3] = make_float4(w * v3.x, w * v3.y, w * v3.z, w * v3.w);
  __syncthreads();

  // ---- reduce over k; lane now acts as channel d (conflict-free: pad 20)
  float f = 0.f;
#pragma unroll
  for (int k = 0; k < Kk; ++k) f += s_mat[grp][k][ln];

  // ---- transpose through LDS so stores are coalesced 64B lines per d-row
  s_out[ln][grp] = f;
  __syncthreads();
  feat[(bg * Dd + (t >> 4)) * Nn + n0 + (t & 15)] = s_out[t >> 4][t & 15];
}

// ------------------------------------------------------------------ launch
extern "C" void kernel_launch(void* const* d_in, const int* in_sizes, int n_in,
                              void* d_out, int out_size, void* d_ws, size_t ws_size,
                              hipStream_t stream) {
  const float* q   = (const float*)d_in[4];   // queryandkey (B,C,N)
  const float* v   = (const float*)d_in[5];   // value       (B,C,N)
  const int*   idx = (const int*)d_in[6];     // idx_knn     (B,N,K)

  float* feat = (float*)d_out;                // (B,C,N)
  float* cent = feat + FEAT;                  // (B,G,N)

  float* qt = (float*)d_ws;                           // (BG,N,d) ~18.9MB
  float* vt = qt + (size_t)BG * Nn * Dd;              // (BG,N,d) ~18.9MB

  zero_cent<<<(CENT + 255) / 256, 256, 0, stream>>>(cent);
  transpose_dn<<<dim3(BG * (Nn / 256), 2), 256, 0, stream>>>(q, v, qt, vt);
  attn_fused<<<dim3(Nn / 16, BG), 256, 0, stream>>>(qt, vt, idx, feat, cent);
}